// CustomLayer_30279519437510
// MI455X (gfx1250) — compile-verified
//
#include <hip/hip_runtime.h>

typedef float v2f __attribute__((ext_vector_type(2)));
typedef float v8f __attribute__((ext_vector_type(8)));

#define NPT    160
#define N3     (160*160*160)      // 4,096,000
#define NLINES (160*160)          // 25,600
#define NLB    32                 // lines per workgroup
#define LSTR   164                // padded dword stride per line (bank-conflict-free, 16B-multiple)
#define LAM    0.05f
#define CG_ITERS 10

// Use CDNA5 async global->LDS copies (GLOBAL_LOAD_ASYNC_TO_LDS_B128) via inline asm;
// the clang builtin's pointer params use unspellable language address spaces.
#define USE_ASYNC_LDS 1

// ---------------- f32 WMMA helper: D = A(16x4) * B(4x16) + C ----------------
static __device__ inline v8f cwmma(v2f a, v2f b, v8f c) {
  // (neg_a, A, neg_b, B, c_mod, C, reuse_a, reuse_b); f32 WMMA: NEG only supports C, keep false.
  return __builtin_amdgcn_wmma_f32_16x16x4_f32(false, a, false, b, (short)0, c, false, false);
}

#if USE_ASYNC_LDS
// Async copy 16 bytes global -> LDS. lds_off = 32-bit LDS byte address (flat ptr truncation,
// which is exactly how addrspacecast flat->local lowers on AMDGPU). Tracked by ASYNCcnt.
static __device__ inline void async_g2l_b128(const float* gsrc, float* ldst) {
  unsigned lds_off = (unsigned)(unsigned long long)(void*)ldst;
  unsigned long long ga = (unsigned long long)(const void*)gsrc;
  asm volatile("global_load_async_to_lds_b128 %0, %1, off"
               :: "v"(lds_off), "v"(ga) : "memory");
}
static __device__ inline void wait_asynccnt0() {
  asm volatile("s_wait_asynccnt 0x0" ::: "memory");
}
#endif

// ---------------- 1D DFT pass (complex GEMM) with transpose --------------------
// Input  layout [P][Q][N], N (=160) fastest: in[L*160 + k],  L = p*160+q
// Output layout [N'][P][Q], Q fastest:       out[n*25600 + L]
// out[n,L] = sum_k (Cr+iCi)[n,k] * in[k,L]
__global__ __launch_bounds__(320) void dft_pass(
    const float* __restrict__ inR, const float* __restrict__ inI,
    float* __restrict__ outR, float* __restrict__ outI,
    const float* __restrict__ Cr, const float* __restrict__ Ci) {
  __shared__ float lR[NLB * LSTR];
  __shared__ float lI[NLB * LSTR];
  const int tid = threadIdx.x;
  const int Lbase = blockIdx.x * NLB;

  // ---- stage 32 complex lines into LDS, line-major [l][k] (padded stride) ----
  {
    const int l  = tid & 31;        // line within tile
    const int g  = tid >> 5;        // 0..9 -> k chunk
    const int k0 = g * 16;
    const float* pr = inR + (size_t)(Lbase + l) * NPT + k0;
    const float* pi = inI + (size_t)(Lbase + l) * NPT + k0;
    float* dr = &lR[l * LSTR + k0];
    float* di = &lI[l * LSTR + k0];
#if USE_ASYNC_LDS
#pragma unroll
    for (int j = 0; j < 16; j += 4) {
      async_g2l_b128(pr + j, dr + j);
      async_g2l_b128(pi + j, di + j);
    }
    wait_asynccnt0();
#else
#pragma unroll
    for (int j = 0; j < 16; j += 4) {
      *(float4*)(dr + j) = *(const float4*)(pr + j);
      *(float4*)(di + j) = *(const float4*)(pi + j);
    }
#endif
  }
  __syncthreads();

  const int wave  = tid >> 5;      // 0..9 == M-tile (rows of DFT matrix)
  const int lane  = tid & 31;
  const int lhalf = lane >> 4;     // 0/1: which half-lane group
  const int lm    = lane & 15;
  const int m     = wave * 16 + lm;       // A-matrix row this lane supplies

  v8f aR0 = {}, aI0 = {}, aR1 = {}, aI1 = {};

#pragma unroll 4
  for (int ks = 0; ks < 40; ++ks) {
    const int kk = ks * 4 + lhalf * 2;    // K slots {0,1} / {2,3} per half
    // A fragments (DFT matrix rows), f32 16x4 layout
    v2f cr  = *(const v2f*)(Cr + (size_t)m * NPT + kk);
    v2f ci  = *(const v2f*)(Ci + (size_t)m * NPT + kk);
    v2f cin = -ci;
    // B fragments: contiguous (X[kk], X[kk+1]) per line -> single ds_load_b64 each
    v2f br0 = *(const v2f*)&lR[lm * LSTR + kk];
    v2f bi0 = *(const v2f*)&lI[lm * LSTR + kk];
    v2f br1 = *(const v2f*)&lR[(16 + lm) * LSTR + kk];
    v2f bi1 = *(const v2f*)&lI[(16 + lm) * LSTR + kk];
    aR0 = cwmma(cr,  br0, aR0);  aR0 = cwmma(cin, bi0, aR0);
    aI0 = cwmma(cr,  bi0, aI0);  aI0 = cwmma(ci,  br0, aI0);
    aR1 = cwmma(cr,  br1, aR1);  aR1 = cwmma(cin, bi1, aR1);
    aI1 = cwmma(cr,  bi1, aI1);  aI1 = cwmma(ci,  br1, aI1);
  }

  // ---- write transposed: out[row*25600 + col] ----
#pragma unroll
  for (int v = 0; v < 8; ++v) {
    const int row = wave * 16 + v + lhalf * 8;
    const size_t o0 = (size_t)row * NLINES + Lbase + lm;
    outR[o0]      = aR0[v];  outI[o0]      = aI0[v];
    outR[o0 + 16] = aR1[v];  outI[o0 + 16] = aI1[v];
  }
}

// ---------------- builders ----------------
__global__ void k_build_mats(float* __restrict__ mats) {
  int idx = blockIdx.x * 256 + threadIdx.x;           // 25600
  int n = idx / NPT, k = idx % NPT;
  int t = (n * k) % NPT;
  float ang = 6.283185307179586f * (float)t / 160.0f;
  float c = cosf(ang), s = sinf(ang);
  mats[idx]               = c;                  // fwd re:  cos(2πnk/N)
  mats[NLINES + idx]      = -s;                 // fwd im: -sin
  mats[2 * NLINES + idx]  = c * (1.0f / 160.0f);// inv re (1/N folded per pass)
  mats[3 * NLINES + idx]  = s * (1.0f / 160.0f);// inv im
}

__global__ void k_build_K(float* __restrict__ K) {
  __shared__ float ctab[160];
  if (threadIdx.x < 160)
    ctab[threadIdx.x] = cosf(6.283185307179586f * (float)threadIdx.x / 160.0f);
  __syncthreads();
  int idx = blockIdx.x * 256 + threadIdx.x;
  int iz = idx % 160, iy = (idx / 160) % 160, ix = idx / NLINES;
  float fx = (float)(ix < 80 ? ix : ix - 160) * (1.0f / 160.0f);
  float fy = (float)(iy < 80 ? iy : iy - 160) * (1.0f / 160.0f);
  float fz = (float)(iz < 80 ? iz : iz - 160) * (1.0f / 160.0f);
  float k2 = fx * fx + fy * fy + fz * fz;
  float D = (1.0f / 3.0f) - ((k2 == 0.0f) ? 0.0f : fz * fz / k2);
  float s1 = 0.f, s2 = 0.f, s3 = 0.f;
  for (int dz = -3; dz <= 3; ++dz)
    for (int dy = -3; dy <= 3; ++dy)
      for (int dx = -3; dx <= 3; ++dx) {
        int d2 = dx * dx + dy * dy + dz * dz;
        if (d2 > 9) continue;
        int t = ix * dx + iy * dy + iz * dz;
        t %= 160; if (t < 0) t += 160;
        float c = ctab[t];
        s3 += c;
        if (d2 <= 4) s2 += c;
        if (d2 <= 1) s1 += c;
      }
  K[idx]          = (1.0f - s1 * (1.0f / 7.0f))   * D;
  K[N3 + idx]     = (1.0f - s2 * (1.0f / 33.0f))  * D;
  K[2 * N3 + idx] = (1.0f - s3 * (1.0f / 123.0f)) * D;
}

__global__ void k_build_mw(const float* __restrict__ x1, const float* __restrict__ x3,
                           float* __restrict__ mw) {
  int i = blockIdx.x * 256 + threadIdx.x;
  float m2 = x3[i] * x3[i];
  mw[i]          = m2 * x1[3 * i + 0];
  mw[N3 + i]     = m2 * x1[3 * i + 1];
  mw[2 * N3 + i] = m2 * x1[3 * i + 2];
}

// ---------------- elementwise ----------------
__global__ void k_copy(const float* __restrict__ s, float* __restrict__ d) {
  int i = blockIdx.x * 256 + threadIdx.x; d[i] = s[i];
}
__global__ void k_pack(const float* __restrict__ z, float* __restrict__ cR, float* __restrict__ cI) {
  int i = blockIdx.x * 256 + threadIdx.x; cR[i] = z[i]; cI[i] = 0.0f;
}
__global__ void k_mulK(const float* __restrict__ Kr, const float* __restrict__ aR,
                       const float* __restrict__ aI, float* __restrict__ oR, float* __restrict__ oI) {
  int i = blockIdx.x * 256 + threadIdx.x; float k = Kr[i]; oR[i] = k * aR[i]; oI[i] = k * aI[i];
}
__global__ void k_vpack(const float* __restrict__ mwr, const float* __restrict__ srcR,
                        float* __restrict__ oR, float* __restrict__ oI) {
  int i = blockIdx.x * 256 + threadIdx.x; oR[i] = mwr[i] * srcR[i]; oI[i] = 0.0f;
}
__global__ void k_accumS(const float* __restrict__ Kr, const float* __restrict__ gR,
                         const float* __restrict__ gI, float* __restrict__ sR,
                         float* __restrict__ sI, int first) {
  int i = blockIdx.x * 256 + threadIdx.x; float k = Kr[i];
  if (first) { sR[i] = k * gR[i]; sI[i] = k * gI[i]; }
  else       { sR[i] += k * gR[i]; sI[i] += k * gI[i]; }
}
__global__ void k_finish(const float* __restrict__ z, const float* __restrict__ re,
                         float* __restrict__ o) {
  int i = blockIdx.x * 256 + threadIdx.x; o[i] = LAM * z[i] + re[i];
}
__global__ void k_init_rp(const float* __restrict__ b, const float* __restrict__ Ap,
                          float* __restrict__ r, float* __restrict__ p) {
  int i = blockIdx.x * 256 + threadIdx.x; float v = b[i] - Ap[i]; r[i] = v; p[i] = v;
}
__global__ void k_update_xr(float* __restrict__ x, float* __restrict__ r,
                            const float* __restrict__ p, const float* __restrict__ Ap,
                            const float* __restrict__ sc) {
  int i = blockIdx.x * 256 + threadIdx.x; float a = sc[3];
  x[i] += a * p[i]; r[i] -= a * Ap[i];
}
__global__ void k_update_p(float* __restrict__ p, const float* __restrict__ r,
                           const float* __restrict__ sc) {
  int i = blockIdx.x * 256 + threadIdx.x; float b = sc[4]; p[i] = r[i] + b * p[i];
}

// ---------------- deterministic reductions ----------------
__global__ void k_dot_partial(const float* __restrict__ a, const float* __restrict__ b,
                              float* __restrict__ parts) {
  __shared__ float sm[256];
  int t = threadIdx.x;
  size_t base = (size_t)blockIdx.x * 2048 + t;
  float s = 0.f;
#pragma unroll
  for (int j = 0; j < 8; ++j) { size_t i = base + (size_t)j * 256; s += a[i] * b[i]; }
  sm[t] = s; __syncthreads();
  for (int off = 128; off > 0; off >>= 1) {
    if (t < off) sm[t] += sm[t + off];
    __syncthreads();
  }
  if (t == 0) parts[blockIdx.x] = sm[0];
}
// mode 0: sc[0]=sum ; mode 1: sc[1]=sum, sc[3]=sc[0]/(sum+eps) ; mode 2: sc[2]=sum, sc[4]=sum/(sc[0]+eps), sc[0]=sum
__global__ void k_reduce_final(const float* __restrict__ parts, int n,
                               float* __restrict__ sc, int mode) {
  __shared__ float sm[256];
  int t = threadIdx.x;
  float s = 0.f;
  for (int i = t; i < n; i += 256) s += parts[i];
  sm[t] = s; __syncthreads();
  for (int off = 128; off > 0; off >>= 1) {
    if (t < off) sm[t] += sm[t + off];
    __syncthreads();
  }
  if (t == 0) {
    float sum = sm[0];
    const float eps = 1e-12f;
    if (mode == 0)      { sc[0] = sum; }
    else if (mode == 1) { sc[1] = sum; sc[3] = sc[0] / (sum + eps); }
    else                { sc[2] = sum; sc[4] = sum / (sc[0] + eps); sc[0] = sum; }
  }
}

// ---------------- host-side orchestration ----------------
#define EW_GRID (N3 / 256)

struct WS {
  float *K, *mw, *FR, *FI, *PR, *PI, *QR, *QI, *SR, *SI;
  float *Rv, *Pv, *Av, *MATS, *parts, *sc;
};

static void pass(const float* ir, const float* ii, float* orr, float* oi,
                 const float* cr, const float* ci, hipStream_t s) {
  dft_pass<<<dim3(NLINES / NLB), dim3(320), 0, s>>>(ir, ii, orr, oi, cr, ci);
}

// Ap = LAM*z + AH(w * A(z)), z and Ap distinct from all scratch complex buffers.
static void applyM(const WS& w, const float* z, float* Ap, hipStream_t s) {
  const float* FWr = w.MATS;
  const float* FWi = w.MATS + NLINES;
  const float* IVr = w.MATS + 2 * NLINES;
  const float* IVi = w.MATS + 3 * NLINES;

  k_pack<<<EW_GRID, 256, 0, s>>>(z, w.PR, w.PI);
  pass(w.PR, w.PI, w.QR, w.QI, FWr, FWi, s);
  pass(w.QR, w.QI, w.PR, w.PI, FWr, FWi, s);
  pass(w.PR, w.PI, w.FR, w.FI, FWr, FWi, s);          // Fz in F

  for (int r = 0; r < 3; ++r) {
    const float* Kr  = w.K  + (size_t)r * N3;
    const float* mwr = w.mw + (size_t)r * N3;
    k_mulK<<<EW_GRID, 256, 0, s>>>(Kr, w.FR, w.FI, w.PR, w.PI);
    pass(w.PR, w.PI, w.QR, w.QI, IVr, IVi, s);
    pass(w.QR, w.QI, w.PR, w.PI, IVr, IVi, s);
    pass(w.PR, w.PI, w.QR, w.QI, IVr, IVi, s);        // image in Q
    k_vpack<<<EW_GRID, 256, 0, s>>>(mwr, w.QR, w.PR, w.PI);
    pass(w.PR, w.PI, w.QR, w.QI, FWr, FWi, s);
    pass(w.QR, w.QI, w.PR, w.PI, FWr, FWi, s);
    pass(w.PR, w.PI, w.QR, w.QI, FWr, FWi, s);        // G_r in Q
    k_accumS<<<EW_GRID, 256, 0, s>>>(Kr, w.QR, w.QI, w.SR, w.SI, r == 0 ? 1 : 0);
  }
  pass(w.SR, w.SI, w.PR, w.PI, IVr, IVi, s);
  pass(w.PR, w.PI, w.QR, w.QI, IVr, IVi, s);
  pass(w.QR, w.QI, w.PR, w.PI, IVr, IVi, s);          // IFFT(S) in P
  k_finish<<<EW_GRID, 256, 0, s>>>(z, w.PR, Ap);
}

static void dot_red(const WS& w, const float* a, const float* b, int mode, hipStream_t s) {
  k_dot_partial<<<dim3(2000), dim3(256), 0, s>>>(a, b, w.parts);
  k_reduce_final<<<dim3(1), dim3(256), 0, s>>>(w.parts, 2000, w.sc, mode);
}

extern "C" void kernel_launch(void* const* d_in, const int* in_sizes, int n_in,
                              void* d_out, int out_size, void* d_ws, size_t ws_size,
                              hipStream_t stream) {
  const float* x1 = (const float*)d_in[0];   // [N3,3] weights
  const float* x2 = (const float*)d_in[1];   // warm start
  const float* x3 = (const float*)d_in[2];   // mask
  const float* bb = (const float*)d_in[3];   // RHS

  float* base = (float*)d_ws;
  WS w;
  size_t o = 0;
  w.K  = base + o; o += (size_t)3 * N3;
  w.mw = base + o; o += (size_t)3 * N3;
  w.FR = base + o; o += N3;  w.FI = base + o; o += N3;
  w.PR = base + o; o += N3;  w.PI = base + o; o += N3;
  w.QR = base + o; o += N3;  w.QI = base + o; o += N3;
  w.SR = base + o; o += N3;  w.SI = base + o; o += N3;
  w.Rv = base + o; o += N3;
  w.Pv = base + o; o += N3;
  w.Av = base + o; o += N3;
  w.MATS  = base + o; o += (size_t)4 * NLINES;
  w.parts = base + o; o += 2048;
  w.sc    = base + o; o += 16;

  float* Xv = (float*)d_out;                 // CG iterate lives in d_out

  // build constants (recomputed every call: deterministic, capture-safe)
  k_build_mats<<<dim3(NLINES / 256), dim3(256), 0, stream>>>(w.MATS);
  k_build_K<<<dim3(EW_GRID), dim3(256), 0, stream>>>(w.K);
  k_build_mw<<<dim3(EW_GRID), dim3(256), 0, stream>>>(x1, x3, w.mw);

  // x = x2 ; r = b - M(x) ; p = r ; rs = <r,r>
  k_copy<<<EW_GRID, 256, 0, stream>>>(x2, Xv);
  applyM(w, Xv, w.Av, stream);
  k_init_rp<<<EW_GRID, 256, 0, stream>>>(bb, w.Av, w.Rv, w.Pv);
  dot_red(w, w.Rv, w.Rv, 0, stream);

  for (int it = 0; it < CG_ITERS; ++it) {
    applyM(w, w.Pv, w.Av, stream);
    dot_red(w, w.Pv, w.Av, 1, stream);                         // alpha = rs/(pAp+eps)
    k_update_xr<<<EW_GRID, 256, 0, stream>>>(Xv, w.Rv, w.Pv, w.Av, w.sc);
    dot_red(w, w.Rv, w.Rv, 2, stream);                         // beta = rs_new/(rs+eps); rs=rs_new
    k_update_p<<<EW_GRID, 256, 0, stream>>>(w.Pv, w.Rv, w.sc);
  }
}